// OdorClassifier_28449863369262
// MI455X (gfx1250) — compile-verified
//
#include <hip/hip_runtime.h>
#include <math.h>

typedef float v2f __attribute__((ext_vector_type(2)));
typedef float v8f __attribute__((ext_vector_type(8)));

#define NNODES 500000
#define NEDGES 8000000
#define NGRAPHS 25000
#define MPAD 25008            // NGRAPHS rounded up to 16
#define RDIM 175
#define RDIM_P 176            // RU16(175)
#define HIN_DIM 185
#define HIN_LD 188            // RU4(185)
#define FC1 96
#define FC2 63
#define FC2_LD 64
#define NTASK 138

__device__ __forceinline__ void atomicMaxF(float* addr, float val) {
    // correct for mixed signs given init = -inf
    if (val >= 0.f) atomicMax((int*)addr, __float_as_int(val));
    else            atomicMin((unsigned int*)addr, __float_as_uint(val));
}

__global__ void fill_kernel(float* __restrict__ p, float v, long long n) {
    long long i = (long long)blockIdx.x * blockDim.x + threadIdx.x;
    if (i < n) p[i] = v;
}

// pack W[K,N] (row-major) into K-pair-interleaved Wp: Wp[(k/2)*Np + n][2] = W[k][n]
// so a lane's B fragment (two consecutive-K values, same column) is one b64 load.
__global__ void pack_weight_pairs(const float* __restrict__ w, float* __restrict__ wp,
                                  int K, int N, int Kp, int Np) {
    int i = blockIdx.x * blockDim.x + threadIdx.x;    // i over Kp*Np
    if (i >= Kp * Np) return;
    int k = i / Np, n = i - k * Np;
    float v = (k < K && n < N) ? w[k * N + n] : 0.f;
    wp[((k >> 1) * Np + n) * 2 + (k & 1)] = v;
}

__global__ void pack_bias(const float* __restrict__ b, float* __restrict__ bp,
                          int N, int Np) {
    int i = blockIdx.x * blockDim.x + threadIdx.x;
    if (i < Np) bp[i] = (i < N) ? b[i] : 0.f;
}

// D = A @ W + bias (+D if accumulate). One wave32 = one 16x16 tile, K-step 4.
// All hot-loop loads unguarded (padded operands); store guards only if guardedStore.
__global__ void wmma_linear(const float* __restrict__ A, const float* __restrict__ Wp,
                            const float* __restrict__ biasp, float* __restrict__ D,
                            int lda, int Kp, int Np, int ldd,
                            int nColTiles, int nRowTiles, int accumulate,
                            int guardedStore, int storeRows, int storeCols) {
    int wid  = (int)(((long long)blockIdx.x * blockDim.x + threadIdx.x) >> 5);
    int lane = threadIdx.x & 31;
    int rowTile = wid / nColTiles;
    if (rowTile >= nRowTiles) return;               // wave-uniform exit
    int colTile = wid - rowTile * nColTiles;
    int row0 = rowTile << 4;
    int col  = (colTile << 4) + (lane & 15);
    int half = lane >> 4;                           // 0 or 1

    float bv = biasp[col];
    v8f c;
    if (accumulate) {
#pragma unroll
        for (int i = 0; i < 8; ++i)
            c[i] = bv + D[(long long)(row0 + half * 8 + i) * ldd + col];
    } else {
#pragma unroll
        for (int i = 0; i < 8; ++i) c[i] = bv;
    }

    const v2f* ap = (const v2f*)(A + (long long)(row0 + (lane & 15)) * lda) + half;
    const v2f* wp = (const v2f*)Wp + (long long)half * Np + col;
    long long wstep = 2 * (long long)Np;            // 2 K-pairs per k0 step

#pragma unroll 2
    for (int k0 = 0; k0 < Kp; k0 += 4) {
        v2f a = *ap;                                // A[row][ka], A[row][ka+1]
        v2f b = *wp;                                // W[ka][col], W[ka+1][col]
        c = __builtin_amdgcn_wmma_f32_16x16x4_f32(false, a, false, b, (short)0, c,
                                                  false, false);
        ap += 2;                                    // +4 floats
        wp += wstep;
    }

    float* dp = D + (long long)(row0 + half * 8) * ldd + col;
    if (!guardedStore) {
#pragma unroll
        for (int i = 0; i < 8; ++i) dp[(long long)i * ldd] = c[i];
    } else {
        bool cok = (col < storeCols);
#pragma unroll
        for (int i = 0; i < 8; ++i) {
            int r = row0 + half * 8 + i;
            if (r < storeRows && cok) dp[(long long)i * ldd] = c[i];
        }
    }
}

// per-edge scatter-max over padded dim (pad cols of y are exact zeros)
__global__ void scatter_max_kernel(const float* __restrict__ y,
                                   const int* __restrict__ src,
                                   const int* __restrict__ dst,
                                   float* __restrict__ agg, int ld) {
    int e = blockIdx.x * blockDim.x + threadIdx.x;
    if (e >= NEDGES) return;
    const float* ys = y + (long long)src[e] * ld;
    float* at = agg + (long long)dst[e] * ld;
    for (int d = 0; d < ld; d += 4) {
        float4 v = *(const float4*)(ys + d);        // 16B aligned: ld % 4 == 0
        atomicMaxF(at + d + 0, v.x);
        atomicMaxF(at + d + 1, v.y);
        atomicMaxF(at + d + 2, v.z);
        atomicMaxF(at + d + 3, v.w);
    }
}

// SELU(fixup(agg)) in-place + atomic segment-sum (pad cols stay exact 0)
__global__ void selu_segsum_kernel(float* __restrict__ agg,
                                   const int* __restrict__ batch,
                                   float* __restrict__ seg, int ld) {
    int node = blockIdx.x * blockDim.x + threadIdx.x;
    if (node >= NNODES) return;
    float* a  = agg + (long long)node * ld;
    float* sg = seg + (long long)batch[node] * ld;
    const float alpha = 1.6732632423543772f, scale = 1.0507009873554805f;
    for (int d = 0; d < ld; ++d) {
        float v = a[d];
        if (!isfinite(v)) v = 0.f;                  // empty segments (-inf) -> 0
        v = (v > 0.f) ? scale * v : scale * alpha * expm1f(v);
        a[d] = v;
        atomicAdd(sg + d, v);
    }
}

// hin[MPAD x HIN_LD]: concat(r, mol) with zero row/col padding
__global__ void concat_kernel(const float* __restrict__ r, const float* __restrict__ mol,
                              float* __restrict__ hin) {
    long long i = (long long)blockIdx.x * blockDim.x + threadIdx.x;
    if (i >= (long long)MPAD * HIN_LD) return;
    int g = (int)(i / HIN_LD);
    int j = (int)(i - (long long)g * HIN_LD);
    float v = 0.f;
    if (g < NGRAPHS) {
        if (j < RDIM)          v = r[(long long)g * RDIM_P + j];
        else if (j < HIN_DIM)  v = mol[(long long)g * 10 + (j - RDIM)];
    }
    hin[i] = v;
}

__global__ void bn_stats_kernel(const float* __restrict__ h, float* __restrict__ mu,
                                float* __restrict__ var, int M, int ld) {
    __shared__ float ssum[256];
    __shared__ float ssq[256];
    int c = blockIdx.x;
    float s = 0.f, q = 0.f;
    for (int r = threadIdx.x; r < M; r += blockDim.x) {
        float v = h[(long long)r * ld + c];
        s += v; q += v * v;
    }
    ssum[threadIdx.x] = s; ssq[threadIdx.x] = q;
    __syncthreads();
    for (int o = 128; o > 0; o >>= 1) {
        if (threadIdx.x < o) {
            ssum[threadIdx.x] += ssum[threadIdx.x + o];
            ssq[threadIdx.x]  += ssq[threadIdx.x + o];
        }
        __syncthreads();
    }
    if (threadIdx.x == 0) {
        float m = ssum[0] / (float)M;
        mu[c]  = m;
        var[c] = ssq[0] / (float)M - m * m;
    }
}

__global__ void bn_relu_kernel(float* __restrict__ h, const float* __restrict__ g,
                               const float* __restrict__ b, const float* __restrict__ mu,
                               const float* __restrict__ var, long long n, int N, int ld) {
    long long i = (long long)blockIdx.x * blockDim.x + threadIdx.x;
    if (i >= n) return;
    int row = (int)(i / N);
    int c   = (int)(i - (long long)row * N);
    long long a = (long long)row * ld + c;
    float v = h[a];
    v = g[c] * (v - mu[c]) / sqrtf(var[c] + 1e-5f) + b[c];
    h[a] = (v > 0.f) ? v : 0.f;
}

__global__ void sigmoid_kernel(float* __restrict__ p, long long n) {
    long long i = (long long)blockIdx.x * blockDim.x + threadIdx.x;
    if (i < n) p[i] = 1.f / (1.f + expf(-p[i]));
}

static inline unsigned grid1(long long n, int blk) { return (unsigned)((n + blk - 1) / blk); }

extern "C" void kernel_launch(void* const* d_in, const int* in_sizes, int n_in,
                              void* d_out, int out_size, void* d_ws, size_t ws_size,
                              hipStream_t stream) {
    const float* x     = (const float*)d_in[0];
    const float* mol   = (const float*)d_in[1];
    const int*   eidx  = (const int*)d_in[2];
    const int*   srcI  = eidx;
    const int*   dstI  = eidx + NEDGES;
    const int*   batch = (const int*)d_in[3];
    const float* wg[4] = {(const float*)d_in[4], (const float*)d_in[6],
                          (const float*)d_in[8], (const float*)d_in[10]};
    const float* bg[4] = {(const float*)d_in[5], (const float*)d_in[7],
                          (const float*)d_in[9], (const float*)d_in[11]};
    const float* wr[4] = {(const float*)d_in[12], (const float*)d_in[14],
                          (const float*)d_in[16], (const float*)d_in[18]};
    const float* br[4] = {(const float*)d_in[13], (const float*)d_in[15],
                          (const float*)d_in[17], (const float*)d_in[19]};
    const float* wfc1 = (const float*)d_in[20]; const float* bfc1 = (const float*)d_in[21];
    const float* bn1g = (const float*)d_in[22]; const float* bn1b = (const float*)d_in[23];
    const float* wfc2 = (const float*)d_in[24]; const float* bfc2 = (const float*)d_in[25];
    const float* bn2g = (const float*)d_in[26]; const float* bn2b = (const float*)d_in[27];
    const float* wout = (const float*)d_in[28]; const float* bout = (const float*)d_in[29];

    const int BLK = 256;
    const int gK[4]   = {48, 20, 27, 36};
    const int gKp[4]  = {48, 20, 28, 36};   // lda of layer input (RU4)
    const int gN[4]   = {20, 27, 36, 36};
    const int gLd[4]  = {20, 28, 36, 36};   // row stride of layer output / seg buffers
    const int gNp[4]  = {32, 32, 48, 48};   // RU16(N) = padded weight width

    float* ws = (float*)d_ws;
    const long long BIG = (long long)NNODES * 36;
    float* buf0 = ws;
    float* buf1 = ws + BIG;
    float* ybuf = ws + 2 * BIG;
    float* cur  = ws + 3 * BIG;
    float* s[4];
    for (int l = 0; l < 4; ++l) { s[l] = cur; cur += (long long)MPAD * gLd[l]; }
    float* segEnd = cur;
    // padded (pair-interleaved) weights + padded biases
    float* wgp[4]; float* bgp[4]; float* wrp[4]; float* brp[4];
    for (int l = 0; l < 4; ++l) { wgp[l] = cur; cur += gKp[l] * gNp[l]; bgp[l] = cur; cur += gNp[l]; }
    for (int l = 0; l < 4; ++l) { wrp[l] = cur; cur += gLd[l] * RDIM_P; brp[l] = cur; cur += RDIM_P; }
    float* wfc1p = cur; cur += HIN_LD * FC1;  float* bfc1p = cur; cur += FC1;
    float* wfc2p = cur; cur += FC1 * FC2_LD;  float* bfc2p = cur; cur += FC2_LD;
    float* woutp = cur; cur += FC2_LD * 144;  float* boutp = cur; cur += 144;
    // MLP-stage buffers alias buf0 (free after the 4 GCN layers)
    float* rbuf = buf0;                                   // MPAD x 176
    float* hin  = rbuf + (long long)MPAD * RDIM_P;        // MPAD x 188
    float* h1   = hin  + (long long)MPAD * HIN_LD;        // MPAD x 96
    float* h2   = h1   + (long long)MPAD * FC1;           // MPAD x 64
    float* mu1  = h2   + (long long)MPAD * FC2_LD;
    float* var1 = mu1 + FC1;
    float* mu2  = var1 + FC1;
    float* var2 = mu2 + FC2;

    // ---- pack all weights/biases into padded buffers ----
    for (int l = 0; l < 4; ++l) {
        pack_weight_pairs<<<grid1(gKp[l] * gNp[l], BLK), BLK, 0, stream>>>(
            wg[l], wgp[l], gK[l], gN[l], gKp[l], gNp[l]);
        pack_bias<<<grid1(gNp[l], BLK), BLK, 0, stream>>>(bg[l], bgp[l], gN[l], gNp[l]);
        pack_weight_pairs<<<grid1(gLd[l] * RDIM_P, BLK), BLK, 0, stream>>>(
            wr[l], wrp[l], gN[l], RDIM, gLd[l], RDIM_P);
        pack_bias<<<grid1(RDIM_P, BLK), BLK, 0, stream>>>(br[l], brp[l], RDIM, RDIM_P);
    }
    pack_weight_pairs<<<grid1(HIN_LD * FC1, BLK), BLK, 0, stream>>>(wfc1, wfc1p, HIN_DIM, FC1, HIN_LD, FC1);
    pack_bias<<<grid1(FC1, BLK), BLK, 0, stream>>>(bfc1, bfc1p, FC1, FC1);
    pack_weight_pairs<<<grid1(FC1 * FC2_LD, BLK), BLK, 0, stream>>>(wfc2, wfc2p, FC1, FC2, FC1, FC2_LD);
    pack_bias<<<grid1(FC2_LD, BLK), BLK, 0, stream>>>(bfc2, bfc2p, FC2, FC2_LD);
    pack_weight_pairs<<<grid1(FC2_LD * 144, BLK), BLK, 0, stream>>>(wout, woutp, FC2, NTASK, FC2_LD, 144);
    pack_bias<<<grid1(144, BLK), BLK, 0, stream>>>(bout, boutp, NTASK, 144);

    // zero all segment-sum accumulators (contiguous, includes row padding)
    long long segTot = segEnd - s[0];
    fill_kernel<<<grid1(segTot, BLK), BLK, 0, stream>>>(s[0], 0.f, segTot);

    const int nodeRowTiles  = NNODES / 16;      // 31250
    const int graphRowTiles = MPAD / 16;        // 1563

    const float* layerIn[4] = {x, buf0, buf1, buf0};
    float* layerAgg[4]      = {buf0, buf1, buf0, buf1};

    for (int l = 0; l < 4; ++l) {
        int ld = gLd[l], Np = gNp[l];
        long long na = (long long)NNODES * ld;
        fill_kernel<<<grid1(na, BLK), BLK, 0, stream>>>(layerAgg[l], -INFINITY, na);
        int ct = Np / 16;
        long long waves = (long long)nodeRowTiles * ct;
        // store guarded on cols only when Np > ld (pad cols beyond row stride)
        wmma_linear<<<grid1(waves * 32, BLK), BLK, 0, stream>>>(
            layerIn[l], wgp[l], bgp[l], ybuf,
            gKp[l], gKp[l], Np, ld, ct, nodeRowTiles, 0,
            /*guarded=*/1, NNODES, ld);
        scatter_max_kernel<<<grid1(NEDGES, BLK), BLK, 0, stream>>>(
            ybuf, srcI, dstI, layerAgg[l], ld);
        selu_segsum_kernel<<<grid1(NNODES, BLK), BLK, 0, stream>>>(
            layerAgg[l], batch, s[l], ld);
    }

    // readout: rbuf = sum_l (s_l @ wr_l + br_l)   [MPAD x 176], fully unguarded
    {
        int ct = RDIM_P / 16;
        long long waves = (long long)graphRowTiles * ct;
        for (int l = 0; l < 4; ++l) {
            wmma_linear<<<grid1(waves * 32, BLK), BLK, 0, stream>>>(
                s[l], wrp[l], brp[l], rbuf,
                gLd[l], gLd[l], RDIM_P, RDIM_P, ct, graphRowTiles, l > 0 ? 1 : 0,
                /*guarded=*/0, MPAD, RDIM_P);
        }
    }

    // hin = concat(rbuf, mol) with zero padding
    concat_kernel<<<grid1((long long)MPAD * HIN_LD, BLK), BLK, 0, stream>>>(rbuf, mol, hin);

    // fc1 -> BN -> ReLU
    {
        int ct = FC1 / 16;
        long long waves = (long long)graphRowTiles * ct;
        wmma_linear<<<grid1(waves * 32, BLK), BLK, 0, stream>>>(
            hin, wfc1p, bfc1p, h1, HIN_LD, HIN_LD, FC1, FC1, ct, graphRowTiles, 0,
            /*guarded=*/0, MPAD, FC1);
        bn_stats_kernel<<<FC1, BLK, 0, stream>>>(h1, mu1, var1, NGRAPHS, FC1);
        long long n = (long long)NGRAPHS * FC1;
        bn_relu_kernel<<<grid1(n, BLK), BLK, 0, stream>>>(h1, bn1g, bn1b, mu1, var1, n, FC1, FC1);
    }
    // fc2 -> BN -> ReLU  (h2 ld=64, valid cols 63; col 63 is exact 0 from padding)
    {
        int ct = FC2_LD / 16;
        long long waves = (long long)graphRowTiles * ct;
        wmma_linear<<<grid1(waves * 32, BLK), BLK, 0, stream>>>(
            h1, wfc2p, bfc2p, h2, FC1, FC1, FC2_LD, FC2_LD, ct, graphRowTiles, 0,
            /*guarded=*/0, MPAD, FC2_LD);
        bn_stats_kernel<<<FC2, BLK, 0, stream>>>(h2, mu2, var2, NGRAPHS, FC2_LD);
        long long n = (long long)NGRAPHS * FC2;
        bn_relu_kernel<<<grid1(n, BLK), BLK, 0, stream>>>(h2, bn2g, bn2b, mu2, var2, n, FC2, FC2_LD);
    }
    // out = sigmoid(h2 @ w_out + b_out) -> d_out [25000 x 138] (guarded epilogue)
    {
        float* out = (float*)d_out;
        int ct = 144 / 16;
        long long waves = (long long)graphRowTiles * ct;
        wmma_linear<<<grid1(waves * 32, BLK), BLK, 0, stream>>>(
            h2, woutp, boutp, out, FC2_LD, FC2_LD, 144, NTASK, ct, graphRowTiles, 0,
            /*guarded=*/1, NGRAPHS, NTASK);
        long long n = (long long)NGRAPHS * NTASK;
        sigmoid_kernel<<<grid1(n, BLK), BLK, 0, stream>>>(out, n);
    }
}